// HopfOscillatorModule_23785528885398
// MI455X (gfx1250) — compile-verified
//
#include <hip/hip_runtime.h>
#include <math.h>

typedef __attribute__((ext_vector_type(2))) float v2f;
typedef __attribute__((ext_vector_type(8))) float v8f;

#define NOSC 32
#define ROWS_PER_BLOCK 128
#define LSTR 34   // padded LDS row stride (even -> 8B-aligned float2 loads)

// workspace layout (floats):
//   [0    .. 1023]  Wc  = w_zd * cos(phi_zd)      (row-major [i][j])
//   [1024 .. 2047]  Ws  = w_zd * sin(phi_zd)
//   [2048 .. 3071]  -Ws
//   [3072 .. 3103]  2*pi*V_MAX*sigmoid(v)
//   [3104 .. 3135]  B_MAX*sigmoid(b)
//   [3136 .. 3167]  C_MAX*sigmoid(c)

__device__ __forceinline__ float sigmoidf_fast(float x) {
    return 1.0f / (1.0f + __expf(-x));
}

__global__ void cpg_precompute(const float* __restrict__ v,
                               const float* __restrict__ b,
                               const float* __restrict__ c,
                               const float* __restrict__ w,
                               const float* __restrict__ phi,
                               float* __restrict__ ws) {
    const float TWO_PI = 6.28318530717958647692f;
    int tid = threadIdx.x;
    if (tid < NOSC) {
        ws[3072 + tid] = TWO_PI * 5.0f * sigmoidf_fast(v[tid]);   // 2*pi*v_a
        ws[3104 + tid] = 2.0f  * sigmoidf_fast(b[tid]);           // b_a
        ws[3136 + tid] = 10.0f * sigmoidf_fast(c[tid]);           // c_a
    }
    for (int f = tid; f < NOSC * NOSC; f += blockDim.x) {
        // _zero_diag mapping: result_flat[f] = (f==1023 || f%33==0) ? 0
        //                     : param[(f/33)*32 + (f%33) - 1]
        float wc = 0.0f, wsn = 0.0f;
        if (f != 1023 && (f % 33) != 0) {
            int p = f / 33, q = f % 33;
            float wv = sigmoidf_fast(w[p * NOSC + q - 1]);            // W_MAX = 1
            float pv = TWO_PI * sigmoidf_fast(phi[p * NOSC + q - 1]); // PHI_MAX = 2*pi
            float sp, cp;
            __sincosf(pv, &sp, &cp);
            wc  = wv * cp;
            wsn = wv * sp;
        }
        ws[f]        = wc;
        ws[1024 + f] = wsn;
        ws[2048 + f] = -wsn;
    }
}

__global__ void __launch_bounds__(256)
cpg_main(const float* __restrict__ st,
         const float* __restrict__ wsp,
         float* __restrict__ out) {
    __shared__ __attribute__((aligned(16))) float sh_u[ROWS_PER_BLOCK * LSTR]; // r*sin(psi)
    __shared__ __attribute__((aligned(16))) float sh_t[ROWS_PER_BLOCK * LSTR]; // r*cos(psi)
    __shared__ __attribute__((aligned(16))) float sh_p[ROWS_PER_BLOCK * LSTR]; // psi
    __shared__ __attribute__((aligned(16))) float sh_wc[NOSC * NOSC];
    __shared__ __attribute__((aligned(16))) float sh_ws[NOSC * NOSC];
    __shared__ __attribute__((aligned(16))) float sh_wn[NOSC * NOSC];

    const int tid  = threadIdx.x;
    const int row0 = blockIdx.x * ROWS_PER_BLOCK;

    // coupling matrices -> LDS
    for (int f = tid; f < NOSC * NOSC; f += 256) {
        sh_wc[f] = wsp[f];
        sh_ws[f] = wsp[1024 + f];
        sh_wn[f] = wsp[2048 + f];
    }

    // Staging pass (coalesced) + fused elementwise outputs (r_d, r_d_dot)
    const int j    = tid & 31;   // oscillator column, fixed per thread
    const int rsub = tid >> 5;   // 0..7
    const float ba = wsp[3104 + j];
    const float ca = wsp[3136 + j];
    for (int it = 0; it < 16; ++it) {
        int r    = rsub + it * 8;          // local row 0..127
        int grow = row0 + r;
        const float* rowp = st + (size_t)grow * 96;
        float psi = rowp[j];
        float rr  = rowp[32 + j];
        float rd  = rowp[64 + j];
        float s, co;
        __sincosf(psi, &s, &co);
        sh_p[r * LSTR + j] = psi;
        sh_u[r * LSTR + j] = rr * s;
        sh_t[r * LSTR + j] = rr * co;
        float* orow = out + (size_t)grow * 96;
        orow[32 + j] = rd;                                        // r_d passthrough
        orow[64 + j] = ca * (ca * 0.25f * (ba - rr) - rd);        // r_d_dot
    }
    __syncthreads();

    // Per-wave 16-row WMMA tile:  A = U*Wc^T + T*(-Ws)^T ; B = T*Wc^T + U*Ws^T
    const int wave  = tid >> 5;
    const int lane  = tid & 31;
    const int mrow  = lane & 15;       // A-matrix row within tile
    const int half  = lane >> 4;       // 0: K=k0,k0+1   1: K=k0+2,k0+3
    const int lrowA = wave * 16 + mrow;
    const int icol  = lane & 15;       // B/D column within 16-wide tile

    v8f accA0 = {}, accA1 = {}, accB0 = {}, accB1 = {};

    for (int k0 = 0; k0 < NOSC; k0 += 4) {
        const int ak = k0 + half * 2;  // even -> aligned float2
        v2f uf = *(const v2f*)&sh_u[lrowA * LSTR + ak];
        v2f tf = *(const v2f*)&sh_t[lrowA * LSTR + ak];
        {   // N-tile 0: columns i = 0..15
            const int i = icol;
            v2f wc  = *(const v2f*)&sh_wc[i * NOSC + ak];
            v2f wsv = *(const v2f*)&sh_ws[i * NOSC + ak];
            v2f wn  = *(const v2f*)&sh_wn[i * NOSC + ak];
            accA0 = __builtin_amdgcn_wmma_f32_16x16x4_f32(false, uf, false, wc,  (short)0, accA0, false, false);
            accA0 = __builtin_amdgcn_wmma_f32_16x16x4_f32(false, tf, false, wn,  (short)0, accA0, false, false);
            accB0 = __builtin_amdgcn_wmma_f32_16x16x4_f32(false, tf, false, wc,  (short)0, accB0, false, false);
            accB0 = __builtin_amdgcn_wmma_f32_16x16x4_f32(false, uf, false, wsv, (short)0, accB0, false, false);
        }
        {   // N-tile 1: columns i = 16..31
            const int i = 16 + icol;
            v2f wc  = *(const v2f*)&sh_wc[i * NOSC + ak];
            v2f wsv = *(const v2f*)&sh_ws[i * NOSC + ak];
            v2f wn  = *(const v2f*)&sh_wn[i * NOSC + ak];
            accA1 = __builtin_amdgcn_wmma_f32_16x16x4_f32(false, uf, false, wc,  (short)0, accA1, false, false);
            accA1 = __builtin_amdgcn_wmma_f32_16x16x4_f32(false, tf, false, wn,  (short)0, accA1, false, false);
            accB1 = __builtin_amdgcn_wmma_f32_16x16x4_f32(false, tf, false, wc,  (short)0, accB1, false, false);
            accB1 = __builtin_amdgcn_wmma_f32_16x16x4_f32(false, uf, false, wsv, (short)0, accB1, false, false);
        }
    }

    // Combine: psi_dot = 2*pi*v_a[i] + cos(psi_i)*A - sin(psi_i)*B
    // D layout: vgpr vv, lanes 0-15 -> M=vv, lanes 16-31 -> M=vv+8; N = lane%16
    for (int n = 0; n < 2; ++n) {
        v8f aA = n ? accA1 : accA0;
        v8f aB = n ? accB1 : accB0;
        const int i  = n * 16 + icol;
        const float va = wsp[3072 + i];
        for (int vv = 0; vv < 8; ++vv) {
            int m    = vv + half * 8;
            int lrow = wave * 16 + m;
            float psi = sh_p[lrow * LSTR + i];
            float s, co;
            __sincosf(psi, &s, &co);
            out[(size_t)(row0 + lrow) * 96 + i] = va + co * aA[vv] - s * aB[vv];
        }
    }
}

extern "C" void kernel_launch(void* const* d_in, const int* in_sizes, int n_in,
                              void* d_out, int out_size, void* d_ws, size_t ws_size,
                              hipStream_t stream) {
    const float* states = (const float*)d_in[0];
    const float* v      = (const float*)d_in[1];
    const float* b      = (const float*)d_in[2];
    const float* c      = (const float*)d_in[3];
    const float* w      = (const float*)d_in[4];
    const float* phi    = (const float*)d_in[5];
    float* out = (float*)d_out;
    float* ws  = (float*)d_ws;

    const int Btot = in_sizes[0] / 96;          // 65536
    cpg_precompute<<<1, 256, 0, stream>>>(v, b, c, w, phi, ws);
    cpg_main<<<Btot / ROWS_PER_BLOCK, 256, 0, stream>>>(states, ws, out);
}